// MultiHeadGATLayer_46943992545841
// MI455X (gfx1250) — compile-verified
//
#include <hip/hip_runtime.h>
#include <math.h>

// GAT layer: N_SRC=N_DST=100000, E=1e6, IN=128, OUT=64, H=2.
// z-projection via V_WMMA_F32_16X16X4_F32 (fp32 tensor path, no precision loss);
// edge softmax + scatter via float atomics (L2-resident working set on MI455X).

typedef __attribute__((ext_vector_type(2))) float v2f;
typedef __attribute__((ext_vector_type(8))) float v8f;

#define IN_D   128
#define HOUT   128   // H*OUT
#define OUT_D  64
#define NEG_SLOPE 0.01f

// ---------------- K0: W (2,64,128) -> Wt[d*128 + c], c = h*64+o ----------------
__global__ void gat_transpose_w(const float* __restrict__ W, float* __restrict__ Wt) {
    int i = blockIdx.x * blockDim.x + threadIdx.x;   // i = c*128 + d, coalesced read
    if (i >= HOUT * IN_D) return;
    int c = i >> 7;
    int d = i & 127;
    Wt[d * HOUT + c] = W[i];
}

// ---------------- K1: Z = Hm (N x 128) * Wt (128 x 128) via WMMA f32 ----------------
// block = 256 threads = 8 waves; block computes 16 rows x 128 cols of Z.
// wave w -> column tile [16w, 16w+16). K-loop: 32 x wmma_f32_16x16x4_f32.
__global__ __launch_bounds__(256) void gat_proj_wmma(
    const float* __restrict__ Hm, const float* __restrict__ Wt,
    float* __restrict__ Z, int N)
{
    int wave = threadIdx.x >> 5;
    int lane = threadIdx.x & 31;
    int row0 = blockIdx.x << 4;
    int col0 = wave << 4;
    int lrow = lane & 15;
    int khi  = (lane >> 4) << 1;          // 0 for lanes 0-15, 2 for lanes 16-31

    int r = row0 + lrow;
    if (r >= N) r = N - 1;                // clamp (tail tiles only)
    const float* arow = Hm + (size_t)r * IN_D;

    v8f acc = {};
    #pragma unroll 8
    for (int k0 = 0; k0 < IN_D; k0 += 4) {
        v2f a, b;
        // A 16x4 layout: lane<16 holds K=k0,k0+1 ; lane>=16 holds K=k0+2,k0+3
        a.x = arow[k0 + khi];
        a.y = arow[k0 + khi + 1];
        // B 4x16 layout (mirrors A): row K striped across lanes 0-15 / 16-31
        b.x = Wt[(size_t)(k0 + khi)     * HOUT + col0 + lrow];
        b.y = Wt[(size_t)(k0 + khi + 1) * HOUT + col0 + lrow];
        acc = __builtin_amdgcn_wmma_f32_16x16x4_f32(
                  false, a, false, b, (short)0, acc, false, false);
    }

    // D layout: VGPR i, lanes 0-15 -> M=i, lanes 16-31 -> M=i+8; N = lane%16
    int rb = row0 + ((lane >> 4) << 3);
    float* zout = Z + (size_t)rb * HOUT + col0 + lrow;
    if (row0 + 16 <= N) {
        // full tile: uniform scalar branch, branch-free store clause
        #pragma unroll
        for (int i = 0; i < 8; ++i)
            zout[(size_t)i * HOUT] = acc[i];
    } else {
        // ragged tail (unused when N % 16 == 0)
        #pragma unroll
        for (int i = 0; i < 8; ++i)
            if (rb + i < N) zout[(size_t)i * HOUT] = acc[i];
    }
}

// ---------------- K2: s[n,h] = sum_o z[n,h,o] * a_w[h, aoff+o] ----------------
// one wave per node; wave32 shuffle reduction.
__global__ __launch_bounds__(256) void gat_score(
    const float* __restrict__ Z, const float* __restrict__ a_w,
    float* __restrict__ S, int N, int aoff)
{
    int wave = threadIdx.x >> 5;
    int lane = threadIdx.x & 31;
    int n = blockIdx.x * 8 + wave;
    if (n >= N) return;
    const float* zr = Z + (size_t)n * HOUT;
    float p0 = zr[lane]          * a_w[aoff + lane]
             + zr[lane + 32]     * a_w[aoff + lane + 32];
    float p1 = zr[64 + lane]     * a_w[128 + aoff + lane]
             + zr[96 + lane]     * a_w[128 + aoff + lane + 32];
    #pragma unroll
    for (int off = 16; off > 0; off >>= 1) {
        p0 += __shfl_down(p0, off, 32);
        p1 += __shfl_down(p1, off, 32);
    }
    if (lane == 0) {
        S[(size_t)n * 2 + 0] = p0;
        S[(size_t)n * 2 + 1] = p1;
    }
}

// ---------------- K3: init m=-inf, denom=0, out=0 (every call: graph is replayed) ----
__global__ void gat_init(float* __restrict__ m, float* __restrict__ denom,
                         float* __restrict__ out, int Nd)
{
    int t = blockIdx.x * blockDim.x + threadIdx.x;
    if (t < Nd * OUT_D) out[t] = 0.0f;
    if (t < Nd * 2) { m[t] = -INFINITY; denom[t] = 0.0f; }
}

// float atomic max via signed/unsigned int ordering trick (valid with -inf init)
__device__ __forceinline__ void atomicMaxFloat(float* addr, float val) {
    if (val >= 0.0f)
        atomicMax((int*)addr, __float_as_int(val));
    else
        atomicMin((unsigned int*)addr, (unsigned int)__float_as_int(val));
}

// ---------------- K4: e = leaky_relu(s_src[si]+s_dst[di]); segment max into m ------
__global__ void gat_edge_max(const float* __restrict__ s_src, const float* __restrict__ s_dst,
                             const int* __restrict__ si, const int* __restrict__ di,
                             float* __restrict__ e, float* __restrict__ m, int E)
{
    int t = blockIdx.x * blockDim.x + threadIdx.x;
    if (t >= E) return;
    int s = si[t], d = di[t];
    #pragma unroll
    for (int h = 0; h < 2; ++h) {
        float v = s_src[(size_t)s * 2 + h] + s_dst[(size_t)d * 2 + h];
        v = (v > 0.0f) ? v : NEG_SLOPE * v;
        e[(size_t)t * 2 + h] = v;
        atomicMaxFloat(&m[(size_t)d * 2 + h], v);
    }
}

// ---------------- K5: ex = exp(e - m[di]); denom += ex (overwrite e with ex) -------
__global__ void gat_edge_exp(const int* __restrict__ di, float* __restrict__ e,
                             const float* __restrict__ m, float* __restrict__ denom, int E)
{
    int t = blockIdx.x * blockDim.x + threadIdx.x;
    if (t >= E) return;
    int d = di[t];
    #pragma unroll
    for (int h = 0; h < 2; ++h) {
        float ex = __expf(e[(size_t)t * 2 + h] - m[(size_t)d * 2 + h]);
        e[(size_t)t * 2 + h] = ex;
        atomicAdd(&denom[(size_t)d * 2 + h], ex);
    }
}

// ---------------- K6: out[d,o] += a0*z[s,0,o] + a1*z[s,1,o]  (64 lanes per edge) ---
__global__ __launch_bounds__(256) void gat_edge_scatter(
    const float* __restrict__ ex, const float* __restrict__ denom,
    const float* __restrict__ z_src,
    const int* __restrict__ si, const int* __restrict__ di,
    float* __restrict__ out, int E)
{
    int t = blockIdx.x * 4 + (threadIdx.x >> 6);
    int o = threadIdx.x & 63;
    if (t >= E) return;
    int s = si[t], d = di[t];
    float a0 = ex[(size_t)t * 2 + 0] / denom[(size_t)d * 2 + 0];
    float a1 = ex[(size_t)t * 2 + 1] / denom[(size_t)d * 2 + 1];
    const float* zr = z_src + (size_t)s * HOUT;
    float v = a0 * zr[o] + a1 * zr[64 + o];
    atomicAdd(&out[(size_t)d * OUT_D + o], v);
}

extern "C" void kernel_launch(void* const* d_in, const int* in_sizes, int n_in,
                              void* d_out, int out_size, void* d_ws, size_t ws_size,
                              hipStream_t stream) {
    const float* h_src = (const float*)d_in[0];
    const float* h_dst = (const float*)d_in[1];
    const float* W_src = (const float*)d_in[2];
    const float* W_dst = (const float*)d_in[3];
    const float* a_w   = (const float*)d_in[4];
    const int* src_idx = (const int*)d_in[5];
    const int* dst_idx = (const int*)d_in[6];

    const int Ns = in_sizes[0] / IN_D;
    const int Nd = in_sizes[1] / IN_D;
    const int E  = in_sizes[5];

    // workspace carve-out
    char* ws = (char*)d_ws;
    size_t off = 0;
    auto alloc = [&](size_t bytes) -> void* {
        void* p = ws + off;
        off += (bytes + 255) & ~(size_t)255;
        return p;
    };
    float* wt_src = (float*)alloc((size_t)IN_D * HOUT * 4);
    float* wt_dst = (float*)alloc((size_t)IN_D * HOUT * 4);
    float* z_src  = (float*)alloc((size_t)Ns * HOUT * 4);
    float* z_dst  = (float*)alloc((size_t)Nd * HOUT * 4);
    float* s_src  = (float*)alloc((size_t)Ns * 2 * 4);
    float* s_dst  = (float*)alloc((size_t)Nd * 2 * 4);
    float* mbuf   = (float*)alloc((size_t)Nd * 2 * 4);
    float* dbuf   = (float*)alloc((size_t)Nd * 2 * 4);
    float* ebuf   = (float*)alloc((size_t)E * 2 * 4);

    float* out = (float*)d_out;

    gat_transpose_w<<<(HOUT * IN_D + 255) / 256, 256, 0, stream>>>(W_src, wt_src);
    gat_transpose_w<<<(HOUT * IN_D + 255) / 256, 256, 0, stream>>>(W_dst, wt_dst);

    gat_proj_wmma<<<(Ns + 15) / 16, 256, 0, stream>>>(h_src, wt_src, z_src, Ns);
    gat_proj_wmma<<<(Nd + 15) / 16, 256, 0, stream>>>(h_dst, wt_dst, z_dst, Nd);

    gat_score<<<(Ns + 7) / 8, 256, 0, stream>>>(z_src, a_w, s_src, Ns, 0);
    gat_score<<<(Nd + 7) / 8, 256, 0, stream>>>(z_dst, a_w, s_dst, Nd, OUT_D);

    gat_init<<<((size_t)Nd * OUT_D + 255) / 256, 256, 0, stream>>>(mbuf, dbuf, out, Nd);

    gat_edge_max<<<(E + 255) / 256, 256, 0, stream>>>(s_src, s_dst, src_idx, dst_idx,
                                                      ebuf, mbuf, E);
    gat_edge_exp<<<(E + 255) / 256, 256, 0, stream>>>(dst_idx, ebuf, mbuf, dbuf, E);
    gat_edge_scatter<<<(E + 3) / 4, 256, 0, stream>>>(ebuf, dbuf, z_src,
                                                      src_idx, dst_idx, out, E);
}